// MinArcSoftmaxLoss_74363063763045
// MI455X (gfx1250) — compile-verified
//
#include <hip/hip_runtime.h>
#include <math.h>

#define TPB 512
#define MARGINF 0.5f
#define SF 64.0f
#define ALPHAF 0.1f
#define INV1PA (1.0f / 1.1f)

// ---- CDNA5 async global->LDS path (guarded: falls back to direct loads) ----
#if defined(__HIP_DEVICE_COMPILE__) && __has_builtin(__builtin_amdgcn_global_load_async_to_lds_b128)
#define USE_ASYNC 1
#else
#define USE_ASYNC 0
#endif

#if USE_ASYNC
typedef int v4i __attribute__((ext_vector_type(4)));
typedef __attribute__((address_space(1))) v4i* gptr_v4i;
typedef __attribute__((address_space(3))) v4i* lptr_v4i;

__device__ __forceinline__ void async_copy16(const void* g, void* l) {
  // builtin signature (per hipcc diagnostic): (global v4i*, lds v4i*, imm offset, imm cpol)
  __builtin_amdgcn_global_load_async_to_lds_b128(
      (gptr_v4i)(void*)g, (lptr_v4i)l, 0, 0);
}
#if __has_builtin(__builtin_amdgcn_s_wait_asynccnt)
#define WAIT_ASYNC(n) __builtin_amdgcn_s_wait_asynccnt(n)
#else
#define WAIT_ASYNC(n) asm volatile("s_wait_asynccnt %0" ::"n"(n) : "memory")
#endif
#endif

// Off-label softmax term: exp(S*((ALPHA*cos(theta-delt) + costh)/(1+ALPHA)) - S)
// cos(theta-delt) = costh*cos(delt) + sqrt(1-costh^2)*sin(delt)   (sin(theta)>=0)
__device__ __forceinline__ float term_off(float cj, float ej, float mean, float sdev) {
  float delt = fmaf(sdev, ej, mean);
  float sd   = __sinf(delt);
  float cd   = __cosf(delt);
  float s2   = fmaf(-cj, cj, 1.0f);
  float sth  = sqrtf(fmaxf(s2, 0.0f));
  float cmm  = fmaf(cj, cd, sth * sd);
  float cm   = fmaf(ALPHAF, cmm, cj) * INV1PA;
  return __expf(fmaf(SF, cm, -SF));  // exp(logit - 64), logit <= 64 always
}

// ---- Pass A: mean and unbiased std of pos_theta over B rows (one block) ----
__global__ __launch_bounds__(TPB) void stats_kernel(const float* __restrict__ costh,
                                                    const int* __restrict__ label,
                                                    float* __restrict__ ws, int Bn, int C) {
  __shared__ float red[TPB];
  const int tid = threadIdx.x;

  float s = 0.0f;
  for (int i = tid; i < Bn; i += TPB)
    s += acosf(costh[(size_t)i * (size_t)C + (size_t)label[i]]);
  red[tid] = s;
  __syncthreads();
  for (int off = TPB / 2; off > 0; off >>= 1) {
    if (tid < off) red[tid] += red[tid + off];
    __syncthreads();
  }
  const float mean = red[0] / (float)Bn;
  __syncthreads();  // everyone has read red[0] before reuse

  float q = 0.0f;
  for (int i = tid; i < Bn; i += TPB) {
    float th = acosf(costh[(size_t)i * (size_t)C + (size_t)label[i]]);
    float d = th - mean;
    q = fmaf(d, d, q);
  }
  red[tid] = q;
  __syncthreads();
  for (int off = TPB / 2; off > 0; off >>= 1) {
    if (tid < off) red[tid] += red[tid + off];
    __syncthreads();
  }
  if (tid == 0) {
    ws[0] = mean;
    ws[1] = sqrtf(fmaxf(red[0] / (float)(Bn - 1), 0.0f));  // ddof=1
  }
}

// ---- Pass B: one block per row; single-pass LSE with fixed max = S ----
__global__ __launch_bounds__(TPB) void row_lse_kernel(const float* __restrict__ costh,
                                                      const float* __restrict__ eps,
                                                      const int* __restrict__ label,
                                                      const float* __restrict__ ws,
                                                      float* __restrict__ rowout,
                                                      int C, int C4, int nFull) {
#if USE_ASYNC
  __shared__ float4 bufC[2][TPB];
  __shared__ float4 bufE[2][TPB];
#endif
  __shared__ float red[TPB];

  const int row = blockIdx.x;
  const int tid = threadIdx.x;
  const float mean = ws[0];
  const float sdev = ws[1];
  const size_t base = (size_t)row * (size_t)C;
  const float4* __restrict__ c4 = (const float4*)(costh + base);
  const float4* __restrict__ e4 = (const float4*)(eps + base);

  float acc = 0.0f;

#if USE_ASYNC
  // Double-buffered async streaming: lane-private LDS slots, so only a
  // wave-level s_wait_asynccnt is needed (async loads complete in order).
  if (nFull > 0) {
    async_copy16(&c4[tid], &bufC[0][tid]);
    async_copy16(&e4[tid], &bufE[0][tid]);
    for (int n = 0; n < nFull; ++n) {
      if (n + 1 < nFull) {
        const int idx = (n + 1) * TPB + tid;
        const int nb = (n + 1) & 1;
        async_copy16(&c4[idx], &bufC[nb][tid]);
        async_copy16(&e4[idx], &bufE[nb][tid]);
        WAIT_ASYNC(2);  // newest 2 = prefetch; current tile's 2 are done
      } else {
        WAIT_ASYNC(0);
      }
      const float4 c = bufC[n & 1][tid];
      const float4 e = bufE[n & 1][tid];
      acc += term_off(c.x, e.x, mean, sdev);
      acc += term_off(c.y, e.y, mean, sdev);
      acc += term_off(c.z, e.z, mean, sdev);
      acc += term_off(c.w, e.w, mean, sdev);
    }
  }
  for (int idx = nFull * TPB + tid; idx < C4; idx += TPB) {  // float4 tail
    const float4 c = c4[idx];
    const float4 e = e4[idx];
    acc += term_off(c.x, e.x, mean, sdev);
    acc += term_off(c.y, e.y, mean, sdev);
    acc += term_off(c.z, e.z, mean, sdev);
    acc += term_off(c.w, e.w, mean, sdev);
  }
#else
  for (int idx = tid; idx < C4; idx += TPB) {
    const float4 c = c4[idx];
    const float4 e = e4[idx];
    acc += term_off(c.x, e.x, mean, sdev);
    acc += term_off(c.y, e.y, mean, sdev);
    acc += term_off(c.z, e.z, mean, sdev);
    acc += term_off(c.w, e.w, mean, sdev);
  }
#endif
  for (int j = C4 * 4 + tid; j < C; j += TPB)  // scalar remainder (C%4 != 0)
    acc += term_off(costh[base + j], eps[base + j], mean, sdev);

  red[tid] = acc;
  __syncthreads();
  for (int off = TPB / 2; off > 0; off >>= 1) {
    if (tid < off) red[tid] += red[tid + off];
    __syncthreads();
  }

  if (tid == 0) {
    const int lab = label[row];
    const float cj = costh[base + (size_t)lab];
    const float ej = eps[base + (size_t)lab];
    // remove the off-label term counted at the label column, add margin term
    const float t_off = term_off(cj, ej, mean, sdev);
    const float theta = acosf(fminf(fmaxf(cj, -1.0f), 1.0f));
    const float cpm = cosf(theta + MARGINF);
    const float cm_lab = fmaf(ALPHAF, cj, cpm) * INV1PA;  // delt=0 at label
    const float t_lab = __expf(fmaf(SF, cm_lab, -SF));
    const float Ssum = red[0] - t_off + t_lab;
    // loss_row = logsumexp - logit_label = (S + log(sum)) - S*cm_lab
    rowout[row] = SF + logf(Ssum) - SF * cm_lab;
  }
}

// ---- Pass C: deterministic mean over rows ----
__global__ __launch_bounds__(TPB) void final_kernel(const float* __restrict__ rowout,
                                                    float* __restrict__ out, int Bn) {
  __shared__ float red[TPB];
  const int tid = threadIdx.x;
  float s = 0.0f;
  for (int i = tid; i < Bn; i += TPB) s += rowout[i];
  red[tid] = s;
  __syncthreads();
  for (int off = TPB / 2; off > 0; off >>= 1) {
    if (tid < off) red[tid] += red[tid + off];
    __syncthreads();
  }
  if (tid == 0) out[0] = red[0] / (float)Bn;
}

extern "C" void kernel_launch(void* const* d_in, const int* in_sizes, int n_in,
                              void* d_out, int out_size, void* d_ws, size_t ws_size,
                              hipStream_t stream) {
  const float* costh = (const float*)d_in[0];
  const float* eps = (const float*)d_in[1];
  const int* label = (const int*)d_in[2];

  const int Bn = in_sizes[2];                     // 512
  const int C = in_sizes[0] / (Bn > 0 ? Bn : 1);  // 100000
  const int C4 = C / 4;                           // 25000 float4s per row
  const int nFull = C4 / TPB;                     // full async tiles per row

  float* ws = (float*)d_ws;  // ws[0]=mean, ws[1]=std
  float* rowout = ws + 2;    // per-row losses

  stats_kernel<<<1, TPB, 0, stream>>>(costh, label, ws, Bn, C);
  row_lse_kernel<<<Bn, TPB, 0, stream>>>(costh, eps, label, ws, rowout, C, C4, nFull);
  final_kernel<<<1, TPB, 0, stream>>>(rowout, (float*)d_out, Bn);
}